// DAGCN_32186484916473
// MI455X (gfx1250) — compile-verified
//
#include <hip/hip_runtime.h>
#include <math.h>

// Problem constants (from reference setup_inputs)
#define NN 2048   // nodes
#define BB 64     // batch
#define DD 64     // d_in == d_out
#define EE 16     // embed dim
#define KI 192    // CHEB_K * D_IN

#define TPAD 68   // padded LDS row stride (floats) for 64-wide GEMM tiles:
                  // 68*4 = 272 B keeps each row 16B-aligned for async B128
                  // writes, and gives conflict-free A-fragment column reads.
#define BUFSTRIDE (2 * 64 * TPAD)  // one [S|X] double-buffer slot (floats)
#define WPAD 65   // padded stride for W_n [192][64]
#define APAD 193  // padded stride for A tile [64][192]

typedef __attribute__((ext_vector_type(2))) float v2f;
typedef __attribute__((ext_vector_type(8))) float v8f;

// Async-LDS builtin parameter type (from hipcc diagnostic):
//   int __vector(4) in AS1 (global) / AS3 (LDS), non-const.
typedef int v4i __attribute__((vector_size(16)));
typedef __attribute__((address_space(1))) v4i* gv4i_p;
typedef __attribute__((address_space(3))) v4i* lv4i_p;

#if defined(__has_builtin)
#if __has_builtin(__builtin_amdgcn_global_load_async_to_lds_b128) && \
    __has_builtin(__builtin_amdgcn_s_wait_asynccnt)
#define USE_ASYNC_LDS 1
#endif
#endif
#ifndef USE_ASYNC_LDS
#define USE_ASYNC_LDS 0
#endif

static __device__ __forceinline__ v8f wmma4(v2f a, v2f b, v8f c) {
  // D = A(16x4, f32) * B(4x16, f32) + C(16x16, f32)
  return __builtin_amdgcn_wmma_f32_16x16x4_f32(
      /*neg_a=*/false, a, /*neg_b=*/false, b,
      /*c_mod=*/(short)0, c, /*reuse_a=*/false, /*reuse_b=*/false);
}

static __device__ __forceinline__ v8f zero8() {
  v8f z;
#pragma unroll
  for (int i = 0; i < 8; ++i) z[i] = 0.f;
  return z;
}

// 16-byte global -> LDS copy. Async (ASYNCcnt-tracked, bypasses VGPRs) when
// the CDNA5 builtin is available; synchronous float4 round-trip otherwise.
static __device__ __forceinline__ void copy16(const float* __restrict__ g,
                                              float* l) {
#if USE_ASYNC_LDS
  __builtin_amdgcn_global_load_async_to_lds_b128((gv4i_p)g, (lv4i_p)l, 0, 0);
#else
  const float4 v = *(const float4*)g;
  l[0] = v.x;
  l[1] = v.y;
  l[2] = v.z;
  l[3] = v.w;
#endif
}

static __device__ __forceinline__ void wait_stage_and_barrier() {
#if USE_ASYNC_LDS
  __builtin_amdgcn_s_wait_asynccnt(0);
  asm volatile("" ::: "memory");
#endif
  __syncthreads();
}

// ---------------------------------------------------------------------------
// Kernel 1: ne = LayerNorm(node_embeddings + time_embeddings) * gamma + beta
// ---------------------------------------------------------------------------
__global__ __launch_bounds__(256) void k_layernorm(
    const float* __restrict__ nodeE, const float* __restrict__ timeE,
    const float* __restrict__ gamma, const float* __restrict__ beta,
    float* __restrict__ ne) {
  int n = blockIdx.x * blockDim.x + threadIdx.x;
  if (n >= NN) return;
  float v[EE];
  float mu = 0.f;
#pragma unroll
  for (int d = 0; d < EE; ++d) {
    v[d] = nodeE[n * EE + d] + timeE[d];
    mu += v[d];
  }
  mu *= (1.f / EE);
  float var = 0.f;
#pragma unroll
  for (int d = 0; d < EE; ++d) {
    float t = v[d] - mu;
    var += t * t;
  }
  var *= (1.f / EE);  // biased variance (matches torch LayerNorm)
  float rs = rsqrtf(var + 1e-12f);
#pragma unroll
  for (int d = 0; d < EE; ++d)
    ne[n * EE + d] = (v[d] - mu) * rs * gamma[d] + beta[d];
}

// ---------------------------------------------------------------------------
// Kernel 2: S = softmax(ne @ ne^T, axis=1)
// One block = 16 rows. 8 waves each do 16 column tiles of WMMA f32 16x16x4
// (K = 16 -> 4 k-steps), logits staged in LDS [16][2048], fused row softmax.
// ---------------------------------------------------------------------------
__global__ __launch_bounds__(256) void k_attn(const float* __restrict__ ne,
                                              float* __restrict__ S) {
  extern __shared__ float rowbuf[];  // [16][NN] = 128 KB
  const int lane = threadIdx.x & 31;
  const int wave = threadIdx.x >> 5;  // 0..7
  const int mrow = lane & 15;
  const int half = lane >> 4;
  const int nblk = blockIdx.x;  // 0..127 (row tile)

  // A fragments: rows nblk*16 + mrow of ne, K index = kk*4 + 2*half + v
  v2f afr[4];
#pragma unroll
  for (int kk = 0; kk < 4; ++kk) {
    int d0 = kk * 4 + 2 * half;
    afr[kk].x = ne[(nblk * 16 + mrow) * EE + d0];
    afr[kk].y = ne[(nblk * 16 + mrow) * EE + d0 + 1];
  }

  for (int ct = wave; ct < NN / 16; ct += 8) {
    v8f acc = zero8();
#pragma unroll
    for (int kk = 0; kk < 4; ++kk) {
      int d0 = kk * 4 + 2 * half;
      v2f bfr;  // B[d, m] = ne[col m, d]
      bfr.x = ne[(ct * 16 + mrow) * EE + d0];
      bfr.y = ne[(ct * 16 + mrow) * EE + d0 + 1];
      acc = wmma4(afr[kk], bfr, acc);
    }
#pragma unroll
    for (int v = 0; v < 8; ++v)
      rowbuf[(v + 8 * half) * NN + ct * 16 + mrow] = acc[v];
  }
  __syncthreads();

  // Row softmax: wave w handles rows 2w, 2w+1. Wave-shuffle reductions.
  for (int rr = 0; rr < 2; ++rr) {
    int r = wave * 2 + rr;
    float* row = rowbuf + (size_t)r * NN;
    float mx = -3.402823466e38f;
    for (int i = lane; i < NN; i += 32) mx = fmaxf(mx, row[i]);
#pragma unroll
    for (int off = 16; off > 0; off >>= 1)
      mx = fmaxf(mx, __shfl_xor(mx, off, 32));
    float sum = 0.f;
    for (int i = lane; i < NN; i += 32) {
      float e = expf(row[i] - mx);
      row[i] = e;
      sum += e;
    }
#pragma unroll
    for (int off = 16; off > 0; off >>= 1) sum += __shfl_xor(sum, off, 32);
    float inv = 1.f / sum;
    size_t base = (size_t)(nblk * 16 + r) * NN;
    for (int i = lane; i < NN; i += 32) S[base + i] = row[i] * inv;
  }
}

// ---------------------------------------------------------------------------
// Kernel 3: Y[b,n,c] = alpha * sum_m S[n,m] * Xin[b,m,c] + beta * Xres[b,n,c]
// Block = 64(n) x 64(c) tile for one batch. Double-buffered LDS tiles staged
// with async global->LDS B128 (ASYNCcnt) so stage mk+1 overlaps the 32 WMMAs
// on tile mk. 8 waves x 2 accumulator tiles.
//   y1 = S@x            (alpha=1, beta=0)
//   y2 = 2*S@y1 - x     (alpha=2, beta=-1)
// ---------------------------------------------------------------------------
__global__ __launch_bounds__(256) void k_gemm_sx(
    const float* __restrict__ S, const float* __restrict__ Xin,
    const float* __restrict__ Xres, float* __restrict__ Y, float alpha,
    float beta) {
  extern __shared__ float smem[];
  // layout per buffer slot: [S tile 64*TPAD][X tile 64*TPAD]; two slots.
  const int tid = threadIdx.x;
  const int lane = tid & 31, wave = tid >> 5;
  const int mrow = lane & 15, half = lane >> 4;
  const int nblk = blockIdx.x;  // n tile (64 rows)
  const int bb = blockIdx.y;    // batch
  const size_t xbase = (size_t)bb * NN * DD;

  const int ti = wave >> 1;        // 0..3: output row tile
  const int tj0 = (wave & 1) * 2;  // output col tiles tj0, tj0+1
  v8f acc0 = zero8(), acc1 = zero8();

  auto stage = [&](int buf, int mk) {
    float* Sl = smem + buf * BUFSTRIDE;
    float* Xl = smem + buf * BUFSTRIDE + 64 * TPAD;
    for (int idx = tid; idx < 64 * 16; idx += 256) {
      int r = idx >> 4, q = idx & 15;
      copy16(S + (size_t)(nblk * 64 + r) * NN + mk * 64 + q * 4,
             Sl + r * TPAD + q * 4);
      copy16(Xin + xbase + (size_t)(mk * 64 + r) * DD + q * 4,
             Xl + r * TPAD + q * 4);
    }
  };

  stage(0, 0);
  for (int mk = 0; mk < NN / 64; ++mk) {
    // Drain stage mk; barrier also proves every wave finished computing on
    // buffer (mk+1)&1 (used at iteration mk-1) before we overwrite it.
    wait_stage_and_barrier();
    if (mk + 1 < NN / 64) stage((mk + 1) & 1, mk + 1);  // prefetch next tile
    const float* Sl = smem + (mk & 1) * BUFSTRIDE;
    const float* Xl = smem + (mk & 1) * BUFSTRIDE + 64 * TPAD;
#pragma unroll
    for (int kk = 0; kk < 16; ++kk) {
      int k0 = kk * 4 + 2 * half;
      v2f a;
      a.x = Sl[(ti * 16 + mrow) * TPAD + k0];
      a.y = Sl[(ti * 16 + mrow) * TPAD + k0 + 1];
      v2f b0, b1;
      b0.x = Xl[k0 * TPAD + tj0 * 16 + mrow];
      b0.y = Xl[(k0 + 1) * TPAD + tj0 * 16 + mrow];
      b1.x = Xl[k0 * TPAD + (tj0 + 1) * 16 + mrow];
      b1.y = Xl[(k0 + 1) * TPAD + (tj0 + 1) * 16 + mrow];
      acc0 = wmma4(a, b0, acc0);
      acc1 = wmma4(a, b1, acc1);
    }
  }
#pragma unroll
  for (int v = 0; v < 8; ++v) {
    int nl = ti * 16 + v + 8 * half;
    size_t o0 = xbase + (size_t)(nblk * 64 + nl) * DD + tj0 * 16 + mrow;
    size_t o1 = o0 + 16;
    Y[o0] = alpha * acc0[v] + beta * Xres[o0];
    Y[o1] = alpha * acc1[v] + beta * Xres[o1];
  }
}

// ---------------------------------------------------------------------------
// Kernel 4: per-node final contraction.
//   W_n[ki,o] = sum_d ne[n,d] * wp[d,ki,o]   (built in LDS)
//   out[b,n,o] = sum_ki A[b,ki] * W_n[ki,o] + bias_n[o]
// where A[b, k*64+i] = {x, y1, y2}[b,n,i]. One block per node n.
// ---------------------------------------------------------------------------
__global__ __launch_bounds__(256) void k_final(
    const float* __restrict__ x, const float* __restrict__ y1,
    const float* __restrict__ y2, const float* __restrict__ ne,
    const float* __restrict__ wp, const float* __restrict__ bp,
    float* __restrict__ out) {
  extern __shared__ float smem[];
  float* Wl = smem;              // [KI][WPAD]
  float* Al = Wl + KI * WPAD;    // [64][APAD]
  float* bias = Al + 64 * APAD;  // [64]
  float* nel = bias + 64;        // [16]
  const int tid = threadIdx.x;
  const int lane = tid & 31, wave = tid >> 5;
  const int mrow = lane & 15, half = lane >> 4;
  const int n = blockIdx.x;

  if (tid < EE) nel[tid] = ne[n * EE + tid];
  __syncthreads();

  // Build W_n (coalesced over o) and bias_n
  for (int idx = tid; idx < KI * 64; idx += 256) {
    int ki = idx >> 6, o = idx & 63;
    float s = 0.f;
#pragma unroll
    for (int d = 0; d < EE; ++d)
      s = fmaf(nel[d], wp[d * KI * 64 + ki * 64 + o], s);
    Wl[ki * WPAD + o] = s;
  }
  if (tid < 64) {
    float s = 0.f;
#pragma unroll
    for (int d = 0; d < EE; ++d) s = fmaf(nel[d], bp[d * 64 + tid], s);
    bias[tid] = s;
  }
  // Stage A = [x | y1 | y2] rows for this node across all 64 batches.
  // (manually unrolled per source to avoid a pointer array)
  for (int idx = tid; idx < 64 * 16; idx += 256) {
    int b = idx >> 4, q = idx & 15;
    size_t gbase = (size_t)b * NN * DD + (size_t)n * DD + q * 4;
    float* dst = Al + b * APAD + q * 4;
    const float4 v0 = *(const float4*)(x + gbase);
    dst[0] = v0.x;
    dst[1] = v0.y;
    dst[2] = v0.z;
    dst[3] = v0.w;
    const float4 v1 = *(const float4*)(y1 + gbase);
    dst[64 + 0] = v1.x;
    dst[64 + 1] = v1.y;
    dst[64 + 2] = v1.z;
    dst[64 + 3] = v1.w;
    const float4 v2 = *(const float4*)(y2 + gbase);
    dst[128 + 0] = v2.x;
    dst[128 + 1] = v2.y;
    dst[128 + 2] = v2.z;
    dst[128 + 3] = v2.w;
  }
  __syncthreads();

  const int ti = wave >> 1;
  const int tj0 = (wave & 1) * 2;
  v8f acc0 = zero8(), acc1 = zero8();
#pragma unroll 8
  for (int kk = 0; kk < KI / 4; ++kk) {
    int k0 = kk * 4 + 2 * half;
    v2f a;
    a.x = Al[(ti * 16 + mrow) * APAD + k0];
    a.y = Al[(ti * 16 + mrow) * APAD + k0 + 1];
    v2f b0, b1;
    b0.x = Wl[k0 * WPAD + tj0 * 16 + mrow];
    b0.y = Wl[(k0 + 1) * WPAD + tj0 * 16 + mrow];
    b1.x = Wl[k0 * WPAD + (tj0 + 1) * 16 + mrow];
    b1.y = Wl[(k0 + 1) * WPAD + (tj0 + 1) * 16 + mrow];
    acc0 = wmma4(a, b0, acc0);
    acc1 = wmma4(a, b1, acc1);
  }
#pragma unroll
  for (int v = 0; v < 8; ++v) {
    int b = ti * 16 + v + 8 * half;
    int o0 = tj0 * 16 + mrow, o1 = o0 + 16;
    size_t base = (size_t)b * NN * DD + (size_t)n * DD;
    out[base + o0] = acc0[v] + bias[o0];
    out[base + o1] = acc1[v] + bias[o1];
  }
}

// ---------------------------------------------------------------------------
// Launch. Workspace layout (requires ~81 MB):
//   ne [2048*16] | S [2048*2048] | y1 [64*2048*64] | y2 [64*2048*64]
// ---------------------------------------------------------------------------
extern "C" void kernel_launch(void* const* d_in, const int* in_sizes, int n_in,
                              void* d_out, int out_size, void* d_ws,
                              size_t ws_size, hipStream_t stream) {
  const float* x = (const float*)d_in[0];
  const float* nodeE = (const float*)d_in[1];
  const float* timeE = (const float*)d_in[2];
  const float* wp = (const float*)d_in[3];
  const float* bp = (const float*)d_in[4];
  const float* g = (const float*)d_in[5];
  const float* be = (const float*)d_in[6];
  float* out = (float*)d_out;

  float* ne = (float*)d_ws;
  float* S = ne + (size_t)NN * EE;
  float* Y1 = S + (size_t)NN * NN;
  float* Y2 = Y1 + (size_t)BB * NN * DD;

  k_layernorm<<<NN / 256, 256, 0, stream>>>(nodeE, timeE, g, be, ne);

  k_attn<<<NN / 16, 256, (size_t)16 * NN * sizeof(float), stream>>>(ne, S);

  dim3 gg(NN / 64, BB);
  size_t shg = (size_t)2 * BUFSTRIDE * sizeof(float);
  k_gemm_sx<<<gg, 256, shg, stream>>>(S, x, x, Y1, 1.f, 0.f);    // y1 = S@x
  k_gemm_sx<<<gg, 256, shg, stream>>>(S, Y1, x, Y2, 2.f, -1.f);  // y2 = 2S@y1-x

  size_t shf = (size_t)(KI * WPAD + 64 * APAD + 64 + EE) * sizeof(float);
  k_final<<<NN, 256, shf, stream>>>(x, Y1, Y2, ne, wp, bp, out);
}